// Quanvolution__gen124_65481071395396
// MI455X (gfx1250) — compile-verified
//
#include <hip/hip_runtime.h>
#include <hip/hip_bf16.h>
#include <stdint.h>

typedef __attribute__((ext_vector_type(2))) float v2f;
typedef __attribute__((ext_vector_type(8))) float v8f;

#define NPATCH 196
#define IMGS_PER_BLOCK 64
#define IMG_STRIDE 60                              // floats; padded strip stride (16B aligned)
#define STRIP_FLOATS (IMGS_PER_BLOCK * IMG_STRIDE) // 3840 floats = 15360 B per strip buffer

__global__ __launch_bounds__(128)
void quanv_wmma_async_kernel(const float* __restrict__ x,
                             const float* __restrict__ params,
                             const float* __restrict__ W,
                             const float* __restrict__ b,
                             float* __restrict__ out)
{
    // B operand: only the 10 valid class columns, per (patch, half, col) -> float2 of
    // {W[n][4p+2h], W[n][4p+2h+1]}. Layout sB[p*20 + h*10 + n]: within a patch, half0
    // occupies banks 0..19 and half1 banks 20..39 -> conflict-free for the 20 active lanes.
    __shared__ v2f   sB[NPATCH * 20];            // 31360 B
    __shared__ float sImg[2 * STRIP_FLOATS];     // 30720 B, double-buffered row-pair strips

    const int tid  = threadIdx.x;
    const int wave = tid >> 5;
    const int lane = tid & 31;
    const int half = lane >> 4;
    const int n    = lane & 15;

    const int blockBase = blockIdx.x * IMGS_PER_BLOCK;
    const int imgLocal  = wave * 16 + n;

    // ---- stage compacted B operand into LDS (branch-free) ----
    for (int e = tid; e < NPATCH * 20; e += 128) {
        const int p   = e / 20;
        const int rem = e - p * 20;
        const int hf  = rem / 10;
        const int nn  = rem - hf * 10;
        const int k0  = p * 4 + hf * 2;
        v2f v;
        v.x = W[nn * 784 + k0];
        v.y = W[nn * 784 + k0 + 1];
        sB[e] = v;
    }

    // async global->LDS staging of rows (2r, 2r+1) of all 64 images (7 x B128 per thread)
    auto stage = [&](int nr, float* dst) {
        for (int idx = tid; idx < IMGS_PER_BLOCK * 14; idx += 128) {
            const int img = idx / 14;
            const int f4  = idx - img * 14;
            const float*  gp  = x + (size_t)(blockBase + img) * 784 + nr * 56 + f4 * 4;
            const uint32_t lds = (uint32_t)(uintptr_t)(dst + img * IMG_STRIDE + f4 * 4);
            asm volatile("global_load_async_to_lds_b128 %0, %1, off"
                         :: "v"(lds), "v"(gp) : "memory");
        }
    };

    const float pA = params[2 * half];
    const float pB = params[2 * half + 1];

    v8f acc = {};

    stage(0, sImg);                      // prologue: strip 0 -> buffer 0

    const v2f* sBlane = sB + half * 10 + (n < 10 ? n : 0);

    for (int r = 0; r < 14; ++r) {
        float* cur = sImg + (r & 1) * STRIP_FLOATS;
        if (r < 13) {
            stage(r + 1, sImg + ((r + 1) & 1) * STRIP_FLOATS);
            // 7 newer loads outstanding; async loads complete in order, so <=7 means
            // strip r's data has landed in LDS.
            asm volatile("s_wait_asynccnt 0x7" ::: "memory");
        } else {
            asm volatile("s_wait_asynccnt 0x0" ::: "memory");
        }
        __syncthreads();                 // strip r visible to all waves (also covers sB on r==0)

        const float* rowp = cur + imgLocal * IMG_STRIDE + half * 28;
        #pragma unroll
        for (int c = 0; c < 14; ++c) {
            const int p = r * 14 + c;
            const v2f px = *reinterpret_cast<const v2f*>(rowp + 2 * c);
            const float cA = __cosf(px.x + pA);
            const float cB = __cosf(px.y + pB);
            v2f a; a.x = cA; a.y = cA * cB;            // (q0,q1) / (q2,q3) per lane-half
            v2f bm; bm.x = 0.0f; bm.y = 0.0f;
            if (n < 10) bm = sBlane[p * 20];           // masked read; zero column otherwise
            acc = __builtin_amdgcn_wmma_f32_16x16x4_f32(
                false, a, false, bm, (short)0, acc, false, false);
        }
        __syncthreads();                 // readers done before this buffer is re-filled at r+2
    }

    // ---- bias + log_softmax over 10 valid columns; C layout: VGPR rr -> image rr + 8*half ----
    const float bval = (n < 10) ? b[n] : 0.0f;
    #pragma unroll
    for (int rrow = 0; rrow < 8; ++rrow) {
        float v  = acc[rrow] + bval;
        float vm = (n < 10) ? v : -__builtin_inff();
        #pragma unroll
        for (int off = 8; off; off >>= 1)
            vm = fmaxf(vm, __shfl_xor(vm, off, 16));
        float e = (n < 10) ? __expf(v - vm) : 0.0f;
        float s = e;
        #pragma unroll
        for (int off = 8; off; off >>= 1)
            s += __shfl_xor(s, off, 16);
        const float lsm = (v - vm) - __logf(s);
        const int image = blockBase + wave * 16 + rrow + 8 * half;
        if (n < 10) out[(size_t)image * 10 + n] = lsm;
    }
}

extern "C" void kernel_launch(void* const* d_in, const int* in_sizes, int n_in,
                              void* d_out, int out_size, void* d_ws, size_t ws_size,
                              hipStream_t stream) {
    const float* x      = (const float*)d_in[0];   // (32768,1,28,28)
    const float* params = (const float*)d_in[1];   // (4,)
    const float* W      = (const float*)d_in[2];   // (10,784)
    const float* b      = (const float*)d_in[3];   // (10,)
    float* out          = (float*)d_out;           // (32768,10)

    const int nblocks = 32768 / IMGS_PER_BLOCK;    // 512
    quanv_wmma_async_kernel<<<nblocks, 128, 0, stream>>>(x, params, W, b, out);
}